// Top1Gate_61933428408750
// MI455X (gfx1250) — compile-verified
//
#include <hip/hip_runtime.h>

// ---------------------------------------------------------------------------
// Top-1 MoE gate for MI455X (gfx1250), wave32 + V_WMMA_F32_16X16X4_F32.
//
//   N = 16384 tokens, D = 2048 model dim, E = 64 experts, capacity = 256.
//
// GEMM kernel: 512 blocks x 8 waves = 4096 waves (4x the K-split) so each
// SIMD holds multiple waves and enough loads stay in flight for 23.3 TB/s.
// Each block: 2 token-groups (16 tokens each) x 4 K-splits (512 K each).
// input is still streamed from HBM exactly once.
//
// d_out (float32, flat, 49155 elems):
//   [0]                l_aux
//   [1..16384]         indices1_s (float values)
//   [16385]            capacity (=256)
//   [16386..32769]     locations1_s (float values)
//   [32770..49153]     gates1_s
//   [49154]            E (=64)
// ---------------------------------------------------------------------------

#define MODEL_DIM   2048
#define NUM_EXPERTS 64
#define NUM_TOKENS  16384
#define KSPLIT      4
#define KRANGE      (MODEL_DIM / KSPLIT)        // 512 K per wave
#define KTILE       64
#define LDS_STRIDE  68                          // 64 + pad: conflict-free b64
#define SPLIT_PITCH (NUM_EXPERTS * LDS_STRIDE)  // 4352 floats per split tile
#define NUM_GROUPS  (NUM_TOKENS / 16)           // 1024 token-groups

#define OUT_LAUX  0
#define OUT_IDX   1
#define OUT_CAP   (1 + NUM_TOKENS)              // 16385
#define OUT_LOC   (2 + NUM_TOKENS)              // 16386
#define OUT_GATES (2 + 2 * NUM_TOKENS)          // 32770
#define OUT_E     (2 + 3 * NUM_TOKENS)          // 49154

typedef __attribute__((ext_vector_type(2))) float v2f;
typedef __attribute__((ext_vector_type(8))) float v8f;

// ---------------------------------------------------------------------------
// Kernel 1: fused logits GEMM + K-split reduce + argmax/softmax stats.
// ---------------------------------------------------------------------------
__global__ __launch_bounds__(256) void moe_gate_gemm(
    const float* __restrict__ input,   // [N, D]
    const float* __restrict__ wg,      // [E, D]
    float* __restrict__ out,           // flat f32 output
    int*   __restrict__ idx_ws,        // [N] expert index per token
    float* __restrict__ meWave,        // [NUM_GROUPS, 64] softmax-mass partials
    int*   __restrict__ histWave)      // [NUM_GROUPS, 64] argmax counts
{
    // 4 wg K-chunk tiles (68 KB); reused as the 24 KB reduction buffer.
    __shared__ __align__(16) float smem[KSPLIT * SPLIT_PITCH];

    const int tid   = threadIdx.x;
    const int lane  = tid & 31;
    const int g     = (tid >> 5) & 1;   // token-group within block
    const int s     = tid >> 6;         // K-split 0..3
    const int half  = lane >> 4;        // 0: lanes 0-15, 1: lanes 16-31
    const int l16   = lane & 15;

    const int groupId   = blockIdx.x * 2 + g;      // 0..1023
    const int tokenBase = groupId * 16;

    // A-matrix row pointer: token l16 of the group, this wave's K window.
    const float* arow = input + (size_t)(tokenBase + l16) * MODEL_DIM
                              + s * KRANGE + 2 * half;

    // wg staging: thread (tid&63) stages expert row (tid&63) for split tid>>6.
    const float* wsrc = wg + (size_t)(tid & 63) * MODEL_DIM + (tid >> 6) * KRANGE;
    float*       wdst = &smem[(tid >> 6) * SPLIT_PITCH + (tid & 63) * LDS_STRIDE];

    const float* btile = &smem[s * SPLIT_PITCH];

    v8f acc0 = {}, acc1 = {}, acc2 = {}, acc3 = {};

    for (int c = 0; c < KRANGE / KTILE; ++c) {       // 8 chunk iterations
        __syncthreads();                             // previous chunk consumed
        {
            const float4* s4 = (const float4*)(wsrc + c * KTILE);
            float4*       d4 = (float4*)wdst;
            #pragma unroll
            for (int i = 0; i < KTILE / 4; ++i) d4[i] = s4[i];
        }
        __syncthreads();

        const float* ar = arow + c * KTILE;
        #pragma unroll
        for (int kk = 0; kk < KTILE; kk += 4) {
            const float2 af = *(const float2*)(ar + kk);
            v2f a; a.x = af.x; a.y = af.y;

            const int klocal = kk + 2 * half;
            const float2 b0f = *(const float2*)&btile[(0 * 16 + l16) * LDS_STRIDE + klocal];
            const float2 b1f = *(const float2*)&btile[(1 * 16 + l16) * LDS_STRIDE + klocal];
            const float2 b2f = *(const float2*)&btile[(2 * 16 + l16) * LDS_STRIDE + klocal];
            const float2 b3f = *(const float2*)&btile[(3 * 16 + l16) * LDS_STRIDE + klocal];
            v2f b0; b0.x = b0f.x; b0.y = b0f.y;
            v2f b1; b1.x = b1f.x; b1.y = b1f.y;
            v2f b2; b2.x = b2f.x; b2.y = b2f.y;
            v2f b3; b3.x = b3f.x; b3.y = b3f.y;

            acc0 = __builtin_amdgcn_wmma_f32_16x16x4_f32(false, a, false, b0, (short)0, acc0, false, false);
            acc1 = __builtin_amdgcn_wmma_f32_16x16x4_f32(false, a, false, b1, (short)0, acc1, false, false);
            acc2 = __builtin_amdgcn_wmma_f32_16x16x4_f32(false, a, false, b2, (short)0, acc2, false, false);
            acc3 = __builtin_amdgcn_wmma_f32_16x16x4_f32(false, a, false, b3, (short)0, acc3, false, false);
        }
    }

    // ---- cross-wave K-split reduction through LDS (lane-major, no conflicts)
    __syncthreads();
    if (s > 0) {
        float* buf = &smem[((s - 1) * 2 + g) * 1024];
        #pragma unroll
        for (int i = 0; i < 8; ++i) {
            buf[(i)      * 32 + lane] = acc0[i];
            buf[(i + 8)  * 32 + lane] = acc1[i];
            buf[(i + 16) * 32 + lane] = acc2[i];
            buf[(i + 24) * 32 + lane] = acc3[i];
        }
    }
    __syncthreads();
    if (s != 0) return;

    #pragma unroll
    for (int sp = 0; sp < KSPLIT - 1; ++sp) {
        const float* buf = &smem[(sp * 2 + g) * 1024];
        #pragma unroll
        for (int i = 0; i < 8; ++i) {
            acc0[i] += buf[(i)      * 32 + lane];
            acc1[i] += buf[(i + 8)  * 32 + lane];
            acc2[i] += buf[(i + 16) * 32 + lane];
            acc3[i] += buf[(i + 24) * 32 + lane];
        }
    }

    // ---- fused per-token argmax / softmax (C layout: token = r + 8*half) ---
    float me0 = 0.f, me1 = 0.f, me2 = 0.f, me3 = 0.f;
    int   ce0 = 0,   ce1 = 0,   ce2 = 0,   ce3 = 0;

    #pragma unroll
    for (int r = 0; r < 8; ++r) {
        float v  = acc0[r]; int ei = l16;
        { float cnd = acc1[r]; if (cnd > v) { v = cnd; ei = 16 + l16; } }
        { float cnd = acc2[r]; if (cnd > v) { v = cnd; ei = 32 + l16; } }
        { float cnd = acc3[r]; if (cnd > v) { v = cnd; ei = 48 + l16; } }

        #pragma unroll
        for (int m = 1; m <= 8; m <<= 1) {           // max+argmax in 16-lane half
            float vo = __shfl_xor(v, m, 32);
            int   io = __shfl_xor(ei, m, 32);
            if (vo > v || (vo == v && io < ei)) { v = vo; ei = io; }
        }

        float p0 = __expf(acc0[r] - v);
        float p1 = __expf(acc1[r] - v);
        float p2 = __expf(acc2[r] - v);
        float p3 = __expf(acc3[r] - v);
        float ps = p0 + p1 + p2 + p3;
        #pragma unroll
        for (int m = 1; m <= 8; m <<= 1) ps += __shfl_xor(ps, m, 32);
        const float invZ = 1.0f / ps;                // == gates1_s at argmax

        me0 += p0 * invZ; me1 += p1 * invZ; me2 += p2 * invZ; me3 += p3 * invZ;
        ce0 += (ei == l16);       ce1 += (ei == 16 + l16);
        ce2 += (ei == 32 + l16);  ce3 += (ei == 48 + l16);

        if (l16 == 0) {
            const int t = tokenBase + r + 8 * half;
            out[OUT_IDX + t]   = (float)ei;
            out[OUT_GATES + t] = invZ;
            idx_ws[t]          = ei;
        }
    }

    me0 += __shfl_xor(me0, 16, 32); me1 += __shfl_xor(me1, 16, 32);
    me2 += __shfl_xor(me2, 16, 32); me3 += __shfl_xor(me3, 16, 32);
    ce0 += __shfl_xor(ce0, 16, 32); ce1 += __shfl_xor(ce1, 16, 32);
    ce2 += __shfl_xor(ce2, 16, 32); ce3 += __shfl_xor(ce3, 16, 32);

    if (half == 0) {
        const int base = groupId * 64 + l16;
        meWave[base]        = me0;  meWave[base + 16]   = me1;
        meWave[base + 32]   = me2;  meWave[base + 48]   = me3;
        histWave[base]      = ce0;  histWave[base + 16] = ce1;
        histWave[base + 32] = ce2;  histWave[base + 48] = ce3;
    }
}

// ---------------------------------------------------------------------------
// Kernel 2: per-expert exclusive prefix over the 1024 group histograms,
// plus l_aux = sum_e(me_e * ce_e) * E / N^2 and the scalar outputs.
// ---------------------------------------------------------------------------
__global__ void moe_gate_scan(
    const float* __restrict__ meWave,
    const int*   __restrict__ histWave,
    int*         __restrict__ groupOff,
    float*       __restrict__ out)
{
    __shared__ float red[64];
    const int e = threadIdx.x;

    int   run = 0;
    float me  = 0.0f;
    for (int wb = 0; wb < NUM_GROUPS; ++wb) {
        const int idx = wb * 64 + e;
        groupOff[idx] = run;
        run += histWave[idx];
        me  += meWave[idx];
    }

    red[e] = me * (float)run;     // me_e * ce_e
    __syncthreads();
    for (int sft = 32; sft > 0; sft >>= 1) {
        if (e < sft) red[e] += red[e + sft];
        __syncthreads();
    }
    if (e == 0) {
        const float scale = (float)NUM_EXPERTS /
                            ((float)NUM_TOKENS * (float)NUM_TOKENS);
        out[OUT_LAUX] = red[0] * scale;
        out[OUT_CAP]  = 256.0f;   // capacity_factor * ceil(N/E)
        out[OUT_E]    = (float)NUM_EXPERTS;
    }
}

// ---------------------------------------------------------------------------
// Kernel 3: locations1_s = groupOff[token/16][e] + rank inside own 16-token
// chunk. 64 blocks x 256 threads, thread = token.
// ---------------------------------------------------------------------------
__global__ __launch_bounds__(256) void moe_gate_loc(
    const int* __restrict__ idx_ws,
    const int* __restrict__ groupOff,
    float*     __restrict__ out)
{
    __shared__ int sh[256];
    const int lt = threadIdx.x;
    const int t  = blockIdx.x * 256 + lt;
    const int e  = idx_ws[t];
    sh[lt] = e;
    __syncthreads();

    const int cs = lt & ~15;
    int rank = 0;
    for (int j = cs; j < lt; ++j) rank += (sh[j] == e);

    const int wb = t >> 4;
    out[OUT_LOC + t] = (float)(groupOff[wb * 64 + e] + rank);
}

// ---------------------------------------------------------------------------
extern "C" void kernel_launch(void* const* d_in, const int* in_sizes, int n_in,
                              void* d_out, int out_size, void* d_ws, size_t ws_size,
                              hipStream_t stream) {
    (void)in_sizes; (void)n_in; (void)out_size; (void)ws_size;

    const float* input = (const float*)d_in[0];   // [16384, 2048] f32
    const float* wg    = (const float*)d_in[1];   // [64, 2048]    f32
    float*       out   = (float*)d_out;

    char* ws = (char*)d_ws;
    int*   idx_ws   = (int*)   ws;                              //  64 KB
    float* meWave   = (float*) (ws + 65536);                    // 256 KB
    int*   histWave = (int*)   (ws + 65536 + 262144);           // 256 KB
    int*   groupOff = (int*)   (ws + 65536 + 2 * 262144);       // 256 KB

    moe_gate_gemm<<<NUM_GROUPS / 2, 256, 0, stream>>>(input, wg, out,
                                                      idx_ws, meWave, histWave);
    moe_gate_scan<<<1, 64, 0, stream>>>(meWave, histWave, groupOff, out);
    moe_gate_loc<<<NUM_TOKENS / 256, 256, 0, stream>>>(idx_ws, groupOff, out);
}